// Gemma3Attention_9552007266390
// MI455X (gfx1250) — compile-verified
//
#include <hip/hip_runtime.h>
#include <hip/hip_bf16.h>

typedef __attribute__((ext_vector_type(2))) float v2f;
typedef __attribute__((ext_vector_type(8))) float v8f;
typedef __attribute__((ext_vector_type(4))) unsigned int u32x4;
typedef __attribute__((ext_vector_type(8))) int i32x8;
typedef __attribute__((ext_vector_type(4))) int i32x4;

#define T_  4096
#define D_  2560
#define NH_ 8      // query heads
#define KH_ 4      // kv heads
#define H_  256    // head dim
#define QCOLS (NH_*H_)   // 2048
#define KCOLS (KH_*H_)   // 1024
#define WIN 1024

// D = A(16x4) * B(4x16) + C, all f32 (CDNA5 V_WMMA_F32_16X16X4_F32)
static __device__ __forceinline__ v8f wmma_f32(v2f a, v2f b, v8f c) {
    return __builtin_amdgcn_wmma_f32_16x16x4_f32(
        /*neg_a=*/false, a, /*neg_b=*/false, b,
        /*c_mod=*/(short)0, c, /*reuse_a=*/false, /*reuse_b=*/false);
}

// ---------------------------------------------------------------------------
// Tensor Data Mover: 2-D f32 tile (tile_rows x tile_cols) from global (row
// stride row_stride elems) into LDS, with 1-dword LDS padding appended after
// every 2^(pad_code+1) dwords (== one row) to kill bank conflicts.
// D# layout per CDNA5 ISA.  Issued per-wave (EXEC
// ignored), tracked with TENSORcnt.  This toolchain uses the 6-arg builtin.
// ---------------------------------------------------------------------------
static __device__ __forceinline__ void tdm_load_tile(
    void* lds_ptr, const float* gptr,
    unsigned tile_cols, unsigned tile_rows,
    unsigned long long row_stride, unsigned pad_code)
{
    unsigned long long ga = (unsigned long long)(size_t)gptr;
    // generic ptr for LDS objects = {shared aperture, lds offset[31:0]}
    unsigned lo = (unsigned)(size_t)lds_ptr;
    unsigned d0len = (unsigned)row_stride;      // tensor_dim0 (elems)
    unsigned d1len = 1u << 20;                  // tensor_dim1 (ample)
    u32x4 g0;
    g0[0] = 1u;                                  // count=1, is_restore=0
    g0[1] = lo;                                  // lds_addr (bytes)
    g0[2] = (unsigned)ga;                        // global_addr[31:0]
    g0[3] = (unsigned)((ga >> 32) & 0x01ffffffu) // global_addr[56:32]
          | (2u << 30);                          // type = 2 ("image")
    i32x8 g1;
    g1[0] = (int)(0x00020000u                    // data_size = 2 -> 4 bytes
          | (1u << 20)                           // pad_enable
          | (pad_code << 22));                   // pad_interval; pad_amount=1dw
    g1[1] = (int)((d0len & 0xffffu) << 16);                              // dim0 lo16
    g1[2] = (int)(((d0len >> 16) & 0xffffu) | ((d1len & 0xffffu) << 16));// dim0 hi/dim1 lo
    g1[3] = (int)(((d1len >> 16) & 0xffffu) | ((tile_cols & 0xffffu) << 16)); // tile_dim0
    g1[4] = (int)(tile_rows & 0xffffu);          // tile_dim1 (tile_dim2 = 0)
    g1[5] = (int)(unsigned)row_stride;           // tensor_dim0_stride[31:0]
    g1[6] = (int)(unsigned)(row_stride >> 32);   // [47:32]; dim1_stride = 0
    g1[7] = 0;
    i32x4 z4 = {0, 0, 0, 0};
    i32x8 z8 = {0, 0, 0, 0, 0, 0, 0, 0};
    __builtin_amdgcn_tensor_load_to_lds(g0, g1, z4, z4, z8, 0);
}

// ---------------------------------------------------------------------------
// Kernel 1: fused QKV projection.  C[t, c] = sum_d x[t,d] * W[d,c]
// grid = (4096/64, T/16), block = 128 (4 waves, one 16x16 WMMA tile each)
// TDM double-buffered K=32 chunks: wave0 issues next tiles, waits tensorcnt<=2
// col sections: [0,2048)->q(ws), [2048,3072)->k_raw(ws), [3072,4096)->v(d_out)
// ---------------------------------------------------------------------------
__global__ __launch_bounds__(128) void qkv_gemm_kernel(
    const float* __restrict__ x,  const float* __restrict__ wq,
    const float* __restrict__ wk, const float* __restrict__ wv,
    float* __restrict__ out_q, float* __restrict__ out_k, float* __restrict__ out_v)
{
    __shared__ float xs[2][16][33];    // TDM pads each 32-dword row by 1 dword
    __shared__ float wls[2][32][65];   // TDM pads each 64-dword row by 1 dword
    const int tid = threadIdx.x, lane = tid & 31, wave = tid >> 5;
    const int t0 = blockIdx.y * 16;
    const int c64 = blockIdx.x * 64;

    const float* wsrc; float* dst; int stride; int cl0;
    if (c64 < QCOLS)              { wsrc = wq; dst = out_q; stride = QCOLS; cl0 = c64; }
    else if (c64 < QCOLS + KCOLS) { wsrc = wk; dst = out_k; stride = KCOLS; cl0 = c64 - QCOLS; }
    else                          { wsrc = wv; dst = out_v; stride = KCOLS; cl0 = c64 - QCOLS - KCOLS; }

    v8f acc = {};
    const int m  = lane & 15;
    const int hi = (lane >> 4) << 1;     // 0 for lanes 0-15, 2 for lanes 16-31
    const int bc = wave * 16 + m;

    if (wave == 0) {   // prologue: tiles for d0 = 0 into buffer 0
        tdm_load_tile(&xs[0][0][0],  x + (size_t)t0 * D_,      32, 16, D_,     4);
        tdm_load_tile(&wls[0][0][0], wsrc + (size_t)0 * stride + cl0, 64, 32, stride, 5);
    }
    int buf = 0;
    for (int d0 = 0; d0 < D_; d0 += 32) {
        if (wave == 0) {
            if (d0 + 32 < D_) {       // issue next chunk into other buffer
                tdm_load_tile(&xs[buf ^ 1][0][0],  x + (size_t)t0 * D_ + d0 + 32, 32, 16, D_, 4);
                tdm_load_tile(&wls[buf ^ 1][0][0], wsrc + (size_t)(d0 + 32) * stride + cl0, 64, 32, stride, 5);
                __builtin_amdgcn_s_wait_tensorcnt(2);  // current pair done
            } else {
                __builtin_amdgcn_s_wait_tensorcnt(0);
            }
        }
        __syncthreads();
        #pragma unroll
        for (int kk = 0; kk < 8; ++kk) {
            int k0 = kk * 4;
            v2f a, b;
            a.x = xs[buf][m][k0 + hi];      a.y = xs[buf][m][k0 + hi + 1];
            b.x = wls[buf][k0 + hi][bc];    b.y = wls[buf][k0 + hi + 1][bc];
            acc = wmma_f32(a, b, acc);
        }
        __syncthreads();
        buf ^= 1;
    }

    const int col  = cl0 + wave * 16 + m;
    const int rowb = t0 + ((lane >> 4) << 3);
    #pragma unroll
    for (int i = 0; i < 8; ++i)
        dst[(size_t)(rowb + i) * stride + col] = acc[i];
}

// ---------------------------------------------------------------------------
// Kernel 2: per-(t, head) RMS-norm (+ q-scale) + RoPE.
// grid = (T, NH_+KH_), block = 256 (one thread per h)
// ---------------------------------------------------------------------------
__global__ __launch_bounds__(256) void norm_rope_kernel(
    float* __restrict__ qbuf, const float* __restrict__ kraw,
    float* __restrict__ kout, const float* __restrict__ qnw,
    const float* __restrict__ knw, const int* __restrict__ pos)
{
    __shared__ float red[256];
    __shared__ float nv[256];
    const int t = blockIdx.x, hd = blockIdx.y, h = threadIdx.x;
    const bool isq = hd < NH_;
    const float* src = isq ? (qbuf + (size_t)t * QCOLS + hd * H_)
                           : (kraw + (size_t)t * KCOLS + (hd - NH_) * H_);
    float val = src[h];

    red[h] = val * val;
    __syncthreads();
    for (int s = 128; s > 0; s >>= 1) {
        if (h < s) red[h] += red[h + s];
        __syncthreads();
    }
    float r = rsqrtf(red[0] * (1.0f / (float)H_) + 1e-6f);
    float y = val * r * (isq ? qnw[h] : knw[h]);
    if (isq) y *= 0.0625f;               // QUERY_PRE_ATTN_SCALAR^-0.5 = 1/16
    nv[h] = y;
    __syncthreads();

    const int j = h & 127;
    float ang = (float)pos[t] * __expf(-(float)(2 * j) * (1.0f / (float)H_) * 9.210340371976184f);
    float sn, cs;
    __sincosf(ang, &sn, &cs);
    float o = (h < 128) ? (nv[h] * cs - nv[h + 128] * sn)
                        : (nv[h] * cs + nv[h - 128] * sn);

    float* dstp = isq ? (qbuf + (size_t)t * QCOLS + hd * H_)
                      : (kout + (size_t)t * KCOLS + (hd - NH_) * H_);
    dstp[h] = o;
}

// ---------------------------------------------------------------------------
// Kernel 3: GQA sliding-window flash attention.
// grid = (T/16, NH_), block = 256 (8 waves).  Q/K/V tiles via TDM (pad_code 7
// -> row stride 257 dwords).  Each wave: 16x16 scores via WMMA (K=256),
// online softmax via 16-lane shuffles, P x V for its 32-wide h slice.
// ---------------------------------------------------------------------------
__global__ __launch_bounds__(256) void attn_kernel(
    const float* __restrict__ q, const float* __restrict__ kc,
    const float* __restrict__ vc, float* __restrict__ attn)
{
    __shared__ float qs[16][H_ + 1];
    __shared__ float ks[16][H_ + 1];
    __shared__ float vs[16][H_ + 1];
    __shared__ float ps[8][16][17];
    const int tid = threadIdx.x, lane = tid & 31, wave = tid >> 5;
    const int t0 = blockIdx.x * 16;
    const int n  = blockIdx.y;
    const int kvh = n >> 1;               // G = N/K = 2

    if (wave == 0)   // q tile: 16 x 256, row stride QCOLS; waited below
        tdm_load_tile(&qs[0][0], q + (size_t)t0 * QCOLS + n * H_, 256, 16, QCOLS, 7);

    float m_i[8], l_i[8];
    v8f o0 = {}, o1 = {};
    #pragma unroll
    for (int i = 0; i < 8; ++i) { m_i[i] = -3.0e38f; l_i[i] = 0.0f; }

    int lo = t0 - (WIN - 1); if (lo < 0) lo = 0;
    const int sb0 = lo & ~15;
    const int m  = lane & 15;
    const int hi = (lane >> 4) << 1;
    const int rowb = t0 + ((lane >> 4) << 3);
    const int prow = ((lane >> 4) << 3);
    const int hb = wave * 32;

    for (int sb = sb0; sb <= t0; sb += 16) {
        __syncthreads();        // prior readers of ks/vs are done
        if (wave == 0) {
            tdm_load_tile(&ks[0][0], kc + (size_t)sb * KCOLS + kvh * H_, 256, 16, KCOLS, 7);
            tdm_load_tile(&vs[0][0], vc + (size_t)sb * KCOLS + kvh * H_, 256, 16, KCOLS, 7);
            __builtin_amdgcn_s_wait_tensorcnt(0);
        }
        __syncthreads();

        // scores(16x16) = Qtile x Ktile^T, K-dim = 256
        v8f sc = {};
        #pragma unroll 8
        for (int kk = 0; kk < 64; ++kk) {
            int k0 = kk * 4;
            v2f a, b;
            a.x = qs[m][k0 + hi];  a.y = qs[m][k0 + hi + 1];
            b.x = ks[m][k0 + hi];  b.y = ks[m][k0 + hi + 1];   // B[K=h][N=s'] = k[s'][h]
            sc = wmma_f32(a, b, sc);
        }

        const int col = sb + m;
        #pragma unroll
        for (int i = 0; i < 8; ++i) {
            int row = rowb + i;
            bool ok = (col <= row) && (row - col < WIN);
            float s = ok ? sc[i] : -1.0e30f;
            float mx = s;
            mx = fmaxf(mx, __shfl_xor(mx, 1, 16));
            mx = fmaxf(mx, __shfl_xor(mx, 2, 16));
            mx = fmaxf(mx, __shfl_xor(mx, 4, 16));
            mx = fmaxf(mx, __shfl_xor(mx, 8, 16));
            float mnew = fmaxf(m_i[i], mx);
            float scl  = __expf(m_i[i] - mnew);
            float p    = ok ? __expf(s - mnew) : 0.0f;
            float rs = p;
            rs += __shfl_xor(rs, 1, 16);
            rs += __shfl_xor(rs, 2, 16);
            rs += __shfl_xor(rs, 4, 16);
            rs += __shfl_xor(rs, 8, 16);
            l_i[i] = l_i[i] * scl + rs;
            m_i[i] = mnew;
            o0[i] *= scl;
            o1[i] *= scl;
            ps[wave][prow + i][m] = p;        // C-layout -> LDS for A re-layout
        }

        // o += P(16x16) x V(16x32 slice), K-dim = 16 (within-wave LDS, in order)
        #pragma unroll
        for (int kk = 0; kk < 4; ++kk) {
            int k0 = kk * 4;
            v2f a, b0, b1;
            a.x  = ps[wave][m][k0 + hi];       a.y  = ps[wave][m][k0 + hi + 1];
            b0.x = vs[k0 + hi][hb + m];        b0.y = vs[k0 + hi + 1][hb + m];
            b1.x = vs[k0 + hi][hb + 16 + m];   b1.y = vs[k0 + hi + 1][hb + 16 + m];
            o0 = wmma_f32(a, b0, o0);
            o1 = wmma_f32(a, b1, o1);
        }
    }

    #pragma unroll
    for (int i = 0; i < 8; ++i) {
        float inv = 1.0f / l_i[i];
        size_t base = (size_t)(rowb + i) * QCOLS + n * H_;
        attn[base + hb + m]      = o0[i] * inv;
        attn[base + hb + 16 + m] = o1[i] * inv;
    }
}

// ---------------------------------------------------------------------------
// Kernel 4: output projection.  o[t,d] = sum_nh attn[t,nh] * wo[nh,d]
// grid = (D_/64, T/16), block = 128 (4 waves), TDM double-buffered like k1.
// ---------------------------------------------------------------------------
__global__ __launch_bounds__(128) void oproj_kernel(
    const float* __restrict__ attn, const float* __restrict__ wo,
    float* __restrict__ out)
{
    __shared__ float as_[2][16][33];
    __shared__ float wls[2][32][65];
    const int tid = threadIdx.x, lane = tid & 31, wave = tid >> 5;
    const int t0 = blockIdx.y * 16;
    const int c0 = blockIdx.x * 64;

    v8f acc = {};
    const int m  = lane & 15;
    const int hi = (lane >> 4) << 1;
    const int bc = wave * 16 + m;

    if (wave == 0) {
        tdm_load_tile(&as_[0][0][0], attn + (size_t)t0 * QCOLS,      32, 16, QCOLS, 4);
        tdm_load_tile(&wls[0][0][0], wo + (size_t)0 * D_ + c0,       64, 32, D_,    5);
    }
    int buf = 0;
    for (int d0 = 0; d0 < QCOLS; d0 += 32) {
        if (wave == 0) {
            if (d0 + 32 < QCOLS) {
                tdm_load_tile(&as_[buf ^ 1][0][0], attn + (size_t)t0 * QCOLS + d0 + 32, 32, 16, QCOLS, 4);
                tdm_load_tile(&wls[buf ^ 1][0][0], wo + (size_t)(d0 + 32) * D_ + c0,    64, 32, D_,    5);
                __builtin_amdgcn_s_wait_tensorcnt(2);
            } else {
                __builtin_amdgcn_s_wait_tensorcnt(0);
            }
        }
        __syncthreads();
        #pragma unroll
        for (int kk = 0; kk < 8; ++kk) {
            int k0 = kk * 4;
            v2f a, b;
            a.x = as_[buf][m][k0 + hi];     a.y = as_[buf][m][k0 + hi + 1];
            b.x = wls[buf][k0 + hi][bc];    b.y = wls[buf][k0 + hi + 1][bc];
            acc = wmma_f32(a, b, acc);
        }
        __syncthreads();
        buf ^= 1;
    }

    const int col  = c0 + wave * 16 + m;
    const int rowb = t0 + ((lane >> 4) << 3);
    #pragma unroll
    for (int i = 0; i < 8; ++i)
        out[(size_t)(rowb + i) * D_ + col] = acc[i];
}

// ---------------------------------------------------------------------------
extern "C" void kernel_launch(void* const* d_in, const int* in_sizes, int n_in,
                              void* d_out, int out_size, void* d_ws, size_t ws_size,
                              hipStream_t stream) {
    const float* x   = (const float*)d_in[0];
    const float* wq  = (const float*)d_in[1];
    const float* wk  = (const float*)d_in[2];
    const float* wv  = (const float*)d_in[3];
    const float* wo  = (const float*)d_in[4];
    const float* qnw = (const float*)d_in[5];
    const float* knw = (const float*)d_in[6];
    const int*   pos = (const int*)d_in[7];

    float* out = (float*)d_out;
    float* out_k = out;                               // (T,K,H) post norm+rope
    float* out_v = out + (size_t)T_ * KCOLS;          // (T,K,H) raw v
    float* out_o = out + (size_t)2 * T_ * KCOLS;      // (T,D)

    // workspace: q (T*2048) + k_raw (T*1024) + attn (T*2048) = 80 MB
    float* ws_q = (float*)d_ws;
    float* ws_k = ws_q + (size_t)T_ * QCOLS;
    float* ws_a = ws_k + (size_t)T_ * KCOLS;

    qkv_gemm_kernel<<<dim3((QCOLS + 2 * KCOLS) / 64, T_ / 16), 128, 0, stream>>>(
        x, wq, wk, wv, ws_q, ws_k, out_v);
    norm_rope_kernel<<<dim3(T_, NH_ + KH_), 256, 0, stream>>>(
        ws_q, ws_k, out_k, qnw, knw, pos);
    attn_kernel<<<dim3(T_ / 16, NH_), 256, 0, stream>>>(
        ws_q, out_k, out_v, ws_a);
    oproj_kernel<<<dim3(D_ / 64, T_ / 16), 128, 0, stream>>>(
        ws_a, wo, out_o);
}